// HyperbolicAdaptiveGraph_5909875000340
// MI455X (gfx1250) — compile-verified
//
#include <hip/hip_runtime.h>
#include <hip/hip_bf16.h>
#include <math.h>

// ---------------------------------------------------------------------------
// HyperbolicAdaptiveGraph on MI455X (gfx1250, wave32, WMMA)
//   E: [8192, 64] f32,  out: [8192, 8192] f32
//   dot = E @ E^T via V_WMMA_F32_16X16X4_F32 (exact fp32 path; GEMM is tiny,
//   the kernel is output-bandwidth + transcendental bound, and fp32 WMMA
//   avoids cancellation error in n_i + n_j - 2*dot).
// ---------------------------------------------------------------------------

#define N_NODES  8192
#define EMB_DIM  64
#define EPSF     1e-8f

typedef __attribute__((ext_vector_type(2))) float v2f;
typedef __attribute__((ext_vector_type(8))) float v8f;

// Kernel 0: per-row squared norms -> d_ws
__global__ void hag_norms_kernel(const float* __restrict__ emb,
                                 float* __restrict__ norms) {
    int i = blockIdx.x * blockDim.x + threadIdx.x;     // 8192 threads total
    const float4* r = (const float4*)(emb + (size_t)i * EMB_DIM);
    float s = 0.0f;
#pragma unroll
    for (int k = 0; k < EMB_DIM / 4; ++k) {
        float4 v = r[k];
        s += v.x * v.x + v.y * v.y + v.z * v.z + v.w * v.w;
    }
    norms[i] = s;
}

// Kernel 1: one workgroup per 16-row block. 8 waves; each wave owns a 16x16
// output tile per column iteration (128 cols / iter, 64 iters).
__global__ void __launch_bounds__(256)
hag_main_kernel(const float* __restrict__ emb,
                const float* __restrict__ norms,
                float* __restrict__ out) {
    const int tid   = threadIdx.x;
    const int wave  = tid >> 5;          // 0..7
    const int lane  = tid & 31;
    const int ln    = lane & 15;         // position within 16-lane half
    const bool lo16 = lane < 16;
    const int koff  = lo16 ? 0 : 2;      // K sub-offset per WMMA f32 layout
    const int i0    = blockIdx.x * 16;   // row block base

    __shared__ float s_part[8][16];      // per-wave partial row sums
    __shared__ float s_inv[16];          // 1 / (rowsum + eps)

    // ---- A fragment: rows i0..i0+15, all of K=64, per-lane packed --------
    // Lane holds row i0+ln; for WMMA step kk it supplies K = 4*kk+koff, +1.
    float af[32];
    {
        const float* pa = emb + (size_t)(i0 + ln) * EMB_DIM + koff;
#pragma unroll
        for (int kk = 0; kk < 16; ++kk) {
            af[2 * kk]     = pa[kk * 4];
            af[2 * kk + 1] = pa[kk * 4 + 1];
        }
    }

    // Row-side constants for this lane's 8 output rows (m = v + (lo16?0:8)).
    const int rowbase = i0 + (lo16 ? 0 : 8);
    float rn[8], invden[8];
#pragma unroll
    for (int v = 0; v < 8; ++v) {
        float n = norms[rowbase + v];
        rn[v] = n;
        float d = 1.0f - n;
        invden[v] = 1.0f / (d * d + EPSF);
    }

    float rs[8];
#pragma unroll
    for (int v = 0; v < 8; ++v) rs[v] = 0.0f;

    // ---- main loop over column tiles -------------------------------------
    for (int cb = 0; cb < N_NODES / 128; ++cb) {
        const int j0 = cb * 128 + wave * 16;   // this wave's 16 columns

        // prefetch next iteration's B rows (uniform condition, EXEC stays full)
        if (cb + 1 < N_NODES / 128) {
            const int jn = (cb + 1) * 128 + wave * 16;
            __builtin_prefetch(emb + (size_t)(jn + ln) * EMB_DIM, 0, 1);
        }

        // B fragment: by symmetry (C = E E^T), B column n = row j0+n of E,
        // so the per-lane load pattern is identical to A's.
        float bf[32];
        {
            const float* pb = emb + (size_t)(j0 + ln) * EMB_DIM + koff;
#pragma unroll
            for (int kk = 0; kk < 16; ++kk) {
                bf[2 * kk]     = pb[kk * 4];
                bf[2 * kk + 1] = pb[kk * 4 + 1];
            }
        }

        // 16 chained fp32 WMMAs: 16x16 tile, K = 64
        v8f c = {};
#pragma unroll
        for (int kk = 0; kk < 16; ++kk) {
            v2f a, b;
            a[0] = af[2 * kk]; a[1] = af[2 * kk + 1];
            b[0] = bf[2 * kk]; b[1] = bf[2 * kk + 1];
            c = __builtin_amdgcn_wmma_f32_16x16x4_f32(
                    /*neg_a=*/false, a, /*neg_b=*/false, b,
                    /*c_mod=*/(short)0, c,
                    /*reuse_a=*/false, /*reuse_b=*/false);
        }

        // Epilogue: hyperbolic distance -> exp(-dist); C layout:
        // element v of lane L is (row = rowbase+v, col = j0 + (L&15)).
        const float cn = norms[j0 + ln];
        float* po = out + (size_t)rowbase * N_NODES + j0 + ln;
#pragma unroll
        for (int v = 0; v < 8; ++v) {
            float dot = c[v];
            float num = fmaxf(rn[v] + cn - 2.0f * dot, 0.0f);
            float x   = fmaf(2.0f * num, invden[v], 1.0f + EPSF);
            float dist = acoshf(x);
            float a = __expf(-dist);          // CURV == 1
            rs[v] += a;
            po[(size_t)v * N_NODES] = a;      // unnormalized, rescaled in pass 2
        }
    }

    // ---- row-sum reduction ----------------------------------------------
    // Butterfly within each 16-lane half (xor masks < 16 stay in-half).
#pragma unroll
    for (int v = 0; v < 8; ++v) {
#pragma unroll
        for (int off = 1; off < 16; off <<= 1)
            rs[v] += __shfl_xor(rs[v], off, 32);
    }
    if (ln == 0) {
        const int mbase = lo16 ? 0 : 8;
#pragma unroll
        for (int v = 0; v < 8; ++v) s_part[wave][mbase + v] = rs[v];
    }
    __threadfence_block();
    __syncthreads();

    if (tid < 16) {
        float s = EPSF;
#pragma unroll
        for (int w = 0; w < 8; ++w) s += s_part[w][tid];
        s_inv[tid] = 1.0f / s;
    }
    __syncthreads();

    // ---- pass 2: normalize this block's 16 x 8192 stripe (L2-resident) ---
    float4* out4 = (float4*)(out + (size_t)i0 * N_NODES);
    for (int idx = tid; idx < 16 * (N_NODES / 4); idx += 256) {
        int r = idx >> 11;                // N_NODES/4 = 2048 float4 per row
        float inv = s_inv[r];
        float4 t = out4[idx];
        t.x *= inv; t.y *= inv; t.z *= inv; t.w *= inv;
        out4[idx] = t;
    }
}

extern "C" void kernel_launch(void* const* d_in, const int* in_sizes, int n_in,
                              void* d_out, int out_size, void* d_ws, size_t ws_size,
                              hipStream_t stream) {
    const float* emb = (const float*)d_in[0];
    float* out   = (float*)d_out;
    float* norms = (float*)d_ws;          // 8192 floats = 32 KB of scratch

    hag_norms_kernel<<<N_NODES / 256, 256, 0, stream>>>(emb, norms);
    hag_main_kernel<<<N_NODES / 16, 256, 0, stream>>>(emb, norms, out);
}